// GraphConv2_26551487823936
// MI455X (gfx1250) — compile-verified
//
#include <hip/hip_runtime.h>

// ---- problem constants (match reference) ----
#define N_NODES 100000
#define N_EDGES 1600000
#define D       128              // D_IN == D_OUT == 128

typedef __attribute__((ext_vector_type(2))) float v2f;
typedef __attribute__((ext_vector_type(8))) float v8f;

// -------------------------------------------------------------------------
// Kernel 1: h = x @ W via V_WMMA_F32_16X16X4_F32 (exact fp32 path).
// Block = 64 threads = 2 waves. Each wave owns a 16-row strip and computes
// all 128 output columns (8 N-tiles). 100000/16 = 6250 waves exactly ->
// grid = 3125 blocks, EXEC is all-ones for every WMMA.
// -------------------------------------------------------------------------
__global__ void __launch_bounds__(64)
gemm_wmma_f32(const float* __restrict__ x, const float* __restrict__ w,
              float* __restrict__ h) {
  const int lane   = threadIdx.x & 31;
  const int wave   = threadIdx.x >> 5;                 // 0..1
  const int laneLo = lane & 15;                        // 0..15
  const int laneHi = lane >> 4;                        // 0 or 1
  const int m0     = (blockIdx.x * 2 + wave) * 16;     // always < N_NODES

  v8f acc[8];
#pragma unroll
  for (int t = 0; t < 8; ++t) acc[t] = (v8f)0.0f;

  const float* xrow = x + (size_t)(m0 + laneLo) * D;

  for (int kk = 0; kk < D; kk += 4) {
    // A fragment (16x4 f32): contiguous float2 per lane
    const float2 av = *(const float2*)(xrow + kk + 2 * laneHi);
    v2f a; a[0] = av.x; a[1] = av.y;

    const int krow = kk + 2 * laneHi;                  // K row this lane-half supplies
#pragma unroll
    for (int nt = 0; nt < 8; ++nt) {
      const int n0 = nt * 16;
      v2f b;
      b[0] = w[(size_t)(krow    ) * D + n0 + laneLo];
      b[1] = w[(size_t)(krow + 1) * D + n0 + laneLo];
      acc[nt] = __builtin_amdgcn_wmma_f32_16x16x4_f32(
          /*neg_a=*/false, a, /*neg_b=*/false, b,
          /*c_mod=*/(short)0, acc[nt], /*reuse_a=*/false, /*reuse_b=*/false);
    }
  }

  // Store: C/D VGPR i -> row m0+i (lanes 0-15), row m0+8+i (lanes 16-31)
#pragma unroll
  for (int nt = 0; nt < 8; ++nt) {
    const int n0 = nt * 16;
#pragma unroll
    for (int i = 0; i < 8; ++i) {
      const int row = m0 + i + 8 * laneHi;
      h[(size_t)row * D + n0 + laneLo] = acc[nt][i];
    }
  }
}

// -------------------------------------------------------------------------
// Kernel 0: zero the accumulator (harness poisons d_out with 0xAA).
// 12.8M floats = 3.2M float4 -> 12500 blocks x 256, exact.
// -------------------------------------------------------------------------
__global__ void __launch_bounds__(256)
zero_out(float4* __restrict__ out) {
  const size_t i = (size_t)blockIdx.x * blockDim.x + threadIdx.x;
  out[i] = make_float4(0.f, 0.f, 0.f, 0.f);
}

// -------------------------------------------------------------------------
// Kernel 2: edge scatter.  One wave per edge: 32 lanes x float4 = 128 cols.
// h (51 MB) and out (51 MB) both fit in 192 MB L2 -> gathers + fp32 atomics
// are L2-resident. 1.6M edges / 8 waves-per-block = 200000 blocks, exact.
// -------------------------------------------------------------------------
__global__ void __launch_bounds__(256)
spmm_scatter(const float* __restrict__ h, const float* __restrict__ vals,
             const int* __restrict__ rows, const int* __restrict__ cols,
             float* __restrict__ out) {
  const int lane = threadIdx.x & 31;
  const size_t e = (size_t)blockIdx.x * 8 + (threadIdx.x >> 5);

  const float v = vals[e];
  const int   r = rows[e];
  const int   c = cols[e];

  const float4 hv = *(const float4*)(h + (size_t)c * D + lane * 4);
  float* op = out + (size_t)r * D + lane * 4;

  unsafeAtomicAdd(op + 0, v * hv.x);   // global_atomic_add_f32
  unsafeAtomicAdd(op + 1, v * hv.y);
  unsafeAtomicAdd(op + 2, v * hv.z);
  unsafeAtomicAdd(op + 3, v * hv.w);
}

// -------------------------------------------------------------------------
// Kernel 3: in-place ReLU, vectorized float4.
// -------------------------------------------------------------------------
__global__ void __launch_bounds__(256)
relu_inplace(float4* __restrict__ out) {
  const size_t i = (size_t)blockIdx.x * blockDim.x + threadIdx.x;
  float4 v = out[i];
  v.x = fmaxf(v.x, 0.f);
  v.y = fmaxf(v.y, 0.f);
  v.z = fmaxf(v.z, 0.f);
  v.w = fmaxf(v.w, 0.f);
  out[i] = v;
}

extern "C" void kernel_launch(void* const* d_in, const int* in_sizes, int n_in,
                              void* d_out, int out_size, void* d_ws, size_t ws_size,
                              hipStream_t stream) {
  const float* x     = (const float*)d_in[0];   // [N_NODES, D]
  const float* wgt   = (const float*)d_in[1];   // [D, D]
  const float* evals = (const float*)d_in[2];   // [N_EDGES]
  const int*   erows = (const int*)d_in[3];     // [N_EDGES]
  const int*   ecols = (const int*)d_in[4];     // [N_EDGES]
  float*       out   = (float*)d_out;           // [N_NODES, D]
  float*       h     = (float*)d_ws;            // scratch: N_NODES*D fp32 = 51.2 MB

  // zero accumulator: 12.8M floats / 4 / 256 = 12500 blocks (exact)
  zero_out<<<12500, 256, 0, stream>>>((float4*)out);

  // GEMM: 100000/16 = 6250 waves, 2 waves per block -> 3125 blocks (exact)
  gemm_wmma_f32<<<3125, 64, 0, stream>>>(x, wgt, h);

  // SpMM scatter: 1.6M edges / 8 waves per block = 200000 blocks (exact)
  spmm_scatter<<<200000, 256, 0, stream>>>(h, evals, erows, ecols, out);

  // ReLU in place
  relu_inplace<<<12500, 256, 0, stream>>>((float4*)out);
}